// GNN_65893388255397
// MI455X (gfx1250) — compile-verified
//
#include <hip/hip_runtime.h>

typedef __attribute__((ext_vector_type(2))) float v2f;
typedef __attribute__((ext_vector_type(8))) float v8f;

#define DIN 128
#define DH  64
#define DOUT 16

// ---------------- degree / symmetric norm ----------------
__global__ void k_init_deg(float* __restrict__ deg, int N) {
    int i = blockIdx.x * 256 + threadIdx.x;
    if (i < N) deg[i] = 1.0f;                       // self-loop contributes 1
}

__global__ void k_deg_accum(const int* __restrict__ ei, float* __restrict__ deg, int E) {
    int e = blockIdx.x * 256 + threadIdx.x;
    if (e < E) atomicAdd(&deg[ei[E + e]], 1.0f);    // dst row of edge_index
}

__global__ void k_deg_to_dinv(float* __restrict__ deg, int N) {
    int i = blockIdx.x * 256 + threadIdx.x;
    if (i < N) deg[i] = rsqrtf(deg[i]);             // deg >= 1 always
}

__global__ void k_zero(float* __restrict__ p, int n) {
    int i = blockIdx.x * 256 + threadIdx.x;
    if (i < n) p[i] = 0.0f;
}

// ---------------- GEMM1: h1 = x @ W1   (N x 128) @ (128 x 64) ----------------
// 8 wave32 per block; each wave owns a 16-row tile and all 64 output columns.
// V_WMMA_F32_16X16X4_F32, K-loop of 32 steps, 4 accumulators.
__global__ __launch_bounds__(256) void k_gemm1(const float* __restrict__ x,
                                               const float* __restrict__ W,
                                               float* __restrict__ h, int N) {
    __shared__ float sW[DIN * DH];                  // 32 KB of the 320 KB WGP LDS
    for (int i = threadIdx.x; i < DIN * DH / 4; i += 256)
        ((float4*)sW)[i] = ((const float4*)W)[i];
    __syncthreads();

    const int wave = threadIdx.x >> 5;
    const int lane = threadIdx.x & 31;
    const int ntiles = (N + 15) >> 4;
    const int tile = blockIdx.x * 8 + wave;
    if (tile >= ntiles) return;                     // uniform per wave: EXEC stays all-1s

    const int half = lane >> 4;                     // 0: K{0,1}, 1: K{2,3}
    const int l15  = lane & 15;
    int row = tile * 16 + l15; if (row >= N) row = N - 1;
    const int koff = half * 2;
    const float* xr = x + (size_t)row * DIN + koff;

    v8f c0 = {}, c1 = {}, c2 = {}, c3 = {};
    for (int k = 0; k < DIN; k += 4) {
        v2f a = { xr[k], xr[k + 1] };               // A 16x4 fragment (2 VGPRs/lane)
        const float* wb = sW + (k + koff) * DH + l15;
        v2f b0 = { wb[0],  wb[DH + 0]  };           // B 4x16 fragments, col tiles 0..3
        v2f b1 = { wb[16], wb[DH + 16] };
        v2f b2 = { wb[32], wb[DH + 32] };
        v2f b3 = { wb[48], wb[DH + 48] };
        c0 = __builtin_amdgcn_wmma_f32_16x16x4_f32(false, a, false, b0, (short)0, c0, false, false);
        c1 = __builtin_amdgcn_wmma_f32_16x16x4_f32(false, a, false, b1, (short)0, c1, false, false);
        c2 = __builtin_amdgcn_wmma_f32_16x16x4_f32(false, a, false, b2, (short)0, c2, false, false);
        c3 = __builtin_amdgcn_wmma_f32_16x16x4_f32(false, a, false, b3, (short)0, c3, false, false);
    }

    const int mbase = half * 8;                     // C/D layout: vgpr i -> M=i (+8 for hi lanes)
    for (int i = 0; i < 8; ++i) {
        int r = tile * 16 + mbase + i;
        if (r < N) {
            float* o = h + (size_t)r * DH + l15;
            o[0] = c0[i]; o[16] = c1[i]; o[32] = c2[i]; o[48] = c3[i];
        }
    }
}

// ---------------- layer-1 scatter: agg1[dst] += h1[src] * dinv[src]*dinv[dst] ----------------
__global__ void k_scatter1(const int* __restrict__ ei, const float* __restrict__ h1,
                           const float* __restrict__ dinv, float* __restrict__ agg,
                           int E, int N) {
    int gid = blockIdx.x * 256 + threadIdx.x;
    int total = (E + N) * DH;
    if (gid >= total) return;
    int e   = gid >> 6;                             // DH == 64
    int col = gid & 63;
    int s, d;
    if (e < E) { s = ei[e]; d = ei[E + e]; } else { s = d = e - E; }
    float v = h1[(size_t)s * DH + col] * dinv[s] * dinv[d];
    atomicAdd(&agg[(size_t)d * DH + col], v);
}

__global__ void k_bias_relu(float* __restrict__ a, const float* __restrict__ b, int N) {
    int gid = blockIdx.x * 256 + threadIdx.x;
    if (gid < N * DH) {
        float v = a[gid] + b[gid & (DH - 1)];
        a[gid] = v > 0.0f ? v : 0.0f;
    }
}

// ---------------- GEMM2: h2 = h @ W2   (N x 64) @ (64 x 16) ----------------
__global__ __launch_bounds__(256) void k_gemm2(const float* __restrict__ h,
                                               const float* __restrict__ W,
                                               float* __restrict__ h2, int N) {
    __shared__ float sW[DH * DOUT];
    for (int i = threadIdx.x; i < DH * DOUT / 4; i += 256)
        ((float4*)sW)[i] = ((const float4*)W)[i];
    __syncthreads();

    const int wave = threadIdx.x >> 5;
    const int lane = threadIdx.x & 31;
    const int ntiles = (N + 15) >> 4;
    const int tile = blockIdx.x * 8 + wave;
    if (tile >= ntiles) return;

    const int half = lane >> 4;
    const int l15  = lane & 15;
    int row = tile * 16 + l15; if (row >= N) row = N - 1;
    const int koff = half * 2;
    const float* hr = h + (size_t)row * DH + koff;

    v8f c0 = {};
    for (int k = 0; k < DH; k += 4) {
        v2f a = { hr[k], hr[k + 1] };
        const float* wb = sW + (k + koff) * DOUT + l15;
        v2f b0 = { wb[0], wb[DOUT] };
        c0 = __builtin_amdgcn_wmma_f32_16x16x4_f32(false, a, false, b0, (short)0, c0, false, false);
    }

    const int mbase = half * 8;
    for (int i = 0; i < 8; ++i) {
        int r = tile * 16 + mbase + i;
        if (r < N) h2[(size_t)r * DOUT + l15] = c0[i];
    }
}

// ---------------- output init with bias, then layer-2 scatter straight into d_out ----------------
__global__ void k_init_out(float* __restrict__ out, const float* __restrict__ b, int N) {
    int gid = blockIdx.x * 256 + threadIdx.x;
    if (gid < N * DOUT) out[gid] = b[gid & (DOUT - 1)];
}

__global__ void k_scatter2(const int* __restrict__ ei, const float* __restrict__ h2,
                           const float* __restrict__ dinv, float* __restrict__ out,
                           int E, int N) {
    int gid = blockIdx.x * 256 + threadIdx.x;
    int total = (E + N) * DOUT;
    if (gid >= total) return;
    int e   = gid >> 4;                             // DOUT == 16
    int col = gid & 15;
    int s, d;
    if (e < E) { s = ei[e]; d = ei[E + e]; } else { s = d = e - E; }
    float v = h2[(size_t)s * DOUT + col] * dinv[s] * dinv[d];
    atomicAdd(&out[(size_t)d * DOUT + col], v);
}

extern "C" void kernel_launch(void* const* d_in, const int* in_sizes, int n_in,
                              void* d_out, int out_size, void* d_ws, size_t ws_size,
                              hipStream_t stream) {
    const float* x  = (const float*)d_in[0];
    const int*   ei = (const int*)d_in[1];
    const float* W1 = (const float*)d_in[2];
    const float* b1 = (const float*)d_in[3];
    const float* W2 = (const float*)d_in[4];
    const float* b2 = (const float*)d_in[5];
    float* out = (float*)d_out;

    const int N = in_sizes[0] / DIN;
    const int E = in_sizes[1] / 2;

    // workspace layout (floats): dinv[N] | h1[N*64] | agg1[N*64] | h2[N*16]  (~58 MB)
    float* dinv = (float*)d_ws;
    float* h1   = dinv + N;
    float* agg1 = h1 + (size_t)N * DH;
    float* h2   = agg1 + (size_t)N * DH;

    auto cdiv = [](long long a, long long b) { return (int)((a + b - 1) / b); };

    // symmetric normalization
    k_init_deg   <<<cdiv(N, 256), 256, 0, stream>>>(dinv, N);
    k_deg_accum  <<<cdiv(E, 256), 256, 0, stream>>>(ei, dinv, E);
    k_deg_to_dinv<<<cdiv(N, 256), 256, 0, stream>>>(dinv, N);

    // layer 1
    const int ntiles = cdiv(N, 16);
    k_gemm1    <<<cdiv(ntiles, 8), 256, 0, stream>>>(x, W1, h1, N);
    k_zero     <<<cdiv((long long)N * DH, 256), 256, 0, stream>>>(agg1, N * DH);
    k_scatter1 <<<cdiv((long long)(E + N) * DH, 256), 256, 0, stream>>>(ei, h1, dinv, agg1, E, N);
    k_bias_relu<<<cdiv((long long)N * DH, 256), 256, 0, stream>>>(agg1, b1, N);

    // layer 2
    k_gemm2    <<<cdiv(ntiles, 8), 256, 0, stream>>>(agg1, W2, h2, N);
    k_init_out <<<cdiv((long long)N * DOUT, 256), 256, 0, stream>>>(out, b2, N);
    k_scatter2 <<<cdiv((long long)(E + N) * DOUT, 256), 256, 0, stream>>>(ei, h2, dinv, out, E, N);
}